// SparseMNISTCNN_69518340653076
// MI455X (gfx1250) — compile-verified
//
#include <hip/hip_runtime.h>
#include <hip/hip_bf16.h>

// ---------------------------------------------------------------------------
// SparseMNISTCNN forward for MI455X (gfx1250, wave32).
//   1) conv5x5(valid) + bias + 2x2 maxpool  -> pooled [128,64,62,62]  (WMMA f32 16x16x4 implicit GEMM,
//      async global->LDS staging, A-fragments register-resident)
//   2) boosted global top-20 per sample     -> (idx, origval) x 20    (register top-20 + LDS merge)
//   3) sparse FC1 gather + ReLU             -> h1 [128,512]           (20-nnz gather, prefetched)
//   4) FC2 + log_softmax                    -> out [128,10]           (wave32 shuffle reduce)
// ---------------------------------------------------------------------------

typedef float v2f __attribute__((ext_vector_type(2)));
typedef float v8f __attribute__((ext_vector_type(8)));

#define BATCH   128
#define COUT    64
#define KS      5
#define KPAD    28            // 25 real taps padded to 7 chunks of K=4
#define HIN     128
#define HC      124
#define HP      62
#define NTOT    (COUT * HP * HP)   // 246016
#define NHID    512
#define NCLS    10
#define KWIN    20
#define NEG_BIG (-3.0e38f)

#define USE_ASYNC_LDS 1        // GLOBAL_LOAD_ASYNC_TO_LDS_B32 staging path

// ---------------------------------------------------------------------------
// Kernel 1: fused conv + bias + maxpool via V_WMMA_F32_16X16X4_F32.
// grid = (8 col-blocks, 31 row-quads, 128 batch), block = 256 (8 waves).
// Workgroup tile: all 64 channels x 4 conv rows x 16 conv cols
//   -> 2 pooled rows x 8 pooled cols.
// Wave w = ns*4 + mt computes channel tile [16*mt,+16) for conv rows
//   {h0+ns, h0+ns+2}; A (weight) fragments live in registers across both.
// GEMM per row-tile: A = weights [16ch x 28], B = im2col [28 x 16], 7 WMMAs.
// ---------------------------------------------------------------------------
__global__ __launch_bounds__(256) void conv_pool_wmma(
    const float* __restrict__ x,        // [B,1,128,128]
    const float* __restrict__ conv_w,   // [64,1,5,5]
    const float* __restrict__ conv_b,   // [64]
    float* __restrict__ pooled)         // [B,64,62,62]
{
    __shared__ float sW[COUT][KPAD];    // padded weights         (7 KB)
    __shared__ float sX[8][20];         // input patch            (0.64 KB)
    __shared__ float sC[16][16][16];    // conv tiles: row*4+mt   (16 KB)

    const int b  = blockIdx.z;
    const int by = blockIdx.y;          // row quad -> pooled rows 2*by, 2*by+1
    const int cb = blockIdx.x;          // 16-wide conv col block
    const int h0 = 4 * by;
    const int w0 = 16 * cb;

    const int tid  = threadIdx.x;
    const int wv   = tid >> 5;          // 0..7
    const int lane = tid & 31;
    const int mt   = wv & 3;            // channel tile
    const int ns   = wv >> 2;           // row slot: rows ns and ns+2
    const int c0   = 16 * mt;

    // stage weights (zero-padded K) into LDS
    for (int t = tid; t < COUT * KPAD; t += 256) {
        const int c = t / KPAD, k = t - KPAD * c;
        sW[c][k] = (k < 25) ? conv_w[c * 25 + k] : 0.0f;
    }

    // stage the 8x20 input patch (cols beyond image -> 0, results discarded)
    if (tid < 160) {
        const int rr = tid / 20, cc = tid - 20 * rr;
        const int gc = w0 + cc;
        const int gcc = (gc < HIN) ? gc : (HIN - 1);   // clamp addr in-bounds
        const float* gptr = &x[((size_t)b * HIN + (h0 + rr)) * HIN + gcc];
#if USE_ASYNC_LDS
        const unsigned lds_off = (unsigned)(size_t)&sX[rr][cc];
        asm volatile("global_load_async_to_lds_b32 %0, %1, off"
                     :: "v"(lds_off), "v"((unsigned long long)(size_t)gptr)
                     : "memory");
#else
        sX[rr][cc] = *gptr;
#endif
    }
#if USE_ASYNC_LDS
    asm volatile("s_wait_asynccnt 0x0" ::: "memory");
    if (tid < 160) {                     // zero-patch the clamped OOB columns
        const int rr = tid / 20, cc = tid - 20 * rr;
        if (w0 + cc >= HIN) sX[rr][cc] = 0.0f;
    }
#endif
    __syncthreads();

    // --- implicit GEMM --------------------------------------------------
    const int mrow  = lane & 15;        // A: M row / B: N col for this lane
    const int khalf = (lane >> 4) << 1; // lanes 16-31 carry K+2 / K+3

    // A fragments (weights) once into registers, reused for both row-tiles
    v2f af[7];
    #pragma unroll
    for (int kk = 0; kk < 7; ++kk) {
        const int kb = 4 * kk + khalf;
        af[kk].x = sW[c0 + mrow][kb];
        af[kk].y = sW[c0 + mrow][kb + 1];
    }

    #pragma unroll
    for (int rsel = 0; rsel < 2; ++rsel) {
        const int trow = ns + 2 * rsel;  // conv row h0 + trow

        v8f acc = {};
        #pragma unroll
        for (int kk = 0; kk < 7; ++kk) {
            const int kb = 4 * kk + khalf;
            v2f bf;
            {   // im2col element (K, N=mrow): tap (ky,kx) of conv row h0+trow
                const int K0 = kb;
                const int ky0 = K0 / 5, kx0 = K0 - 5 * ky0;
                bf.x = (K0 < 25) ? sX[trow + ky0][mrow + kx0] : 0.0f;
                const int K1 = kb + 1;
                const int ky1 = K1 / 5, kx1 = K1 - 5 * ky1;
                bf.y = (K1 < 25) ? sX[trow + ky1][mrow + kx1] : 0.0f;
            }
            // D = A(16x4,f32) * B(4x16,f32) + C(16x16,f32)
            acc = __builtin_amdgcn_wmma_f32_16x16x4_f32(
                false, af[kk], false, bf, (short)0, acc, false, false);
        }

        // spill: VGPR j, lane l -> channel (j + 8*(l>=16)), col l%16
        #pragma unroll
        for (int j = 0; j < 8; ++j) {
            const int m = j + ((lane >> 4) << 3);
            sC[trow * 4 + mt][m][lane & 15] = acc[j];
        }
    }
    __syncthreads();

    // --- 2x2 maxpool + bias, write pooled in reference-flatten order -----
    for (int p = tid; p < 1024; p += 256) {
        const int prs = p >> 9;          // pooled row within block (0..1)
        const int rem = p & 511;
        const int ch = rem >> 3, q = rem & 7;
        const int mt2 = ch >> 4, m = ch & 15;
        const int t0 = (2 * prs) * 4 + mt2;
        const int t1 = (2 * prs + 1) * 4 + mt2;
        const float v0 = fmaxf(sC[t0][m][2 * q], sC[t0][m][2 * q + 1]);
        const float v1 = fmaxf(sC[t1][m][2 * q], sC[t1][m][2 * q + 1]);
        const int pc = 8 * cb + q;
        if (pc < HP) {
            const int pr = 2 * by + prs;
            pooled[(((size_t)b * COUT + ch) * HP + pr) * HP + pc] =
                fmaxf(v0, v1) + conv_b[ch];
        }
    }
}

// ---------------------------------------------------------------------------
// Kernel 2: boosted global top-20 per sample. grid = 128, block = 256.
// Phase A: each thread keeps a register-sorted top-20 of its strided slice
//          of boosted values (boost factor per channel precomputed in LDS).
// Phase B: 256-way merge via per-thread cursors + LDS argmax tree, 20 rounds.
// Winners record (flat index, ORIGINAL pooled value) per the reference mask.
// ---------------------------------------------------------------------------
__global__ __launch_bounds__(256) void topk_boosted(
    const float* __restrict__ pooled,
    const float* __restrict__ duty,     // [64]
    int*   __restrict__ widx,           // [B,20]
    float* __restrict__ wval)           // [B,20]
{
    __shared__ float sFac[COUT];
    __shared__ float sVal[256 * KWIN];
    __shared__ int   sIdx[256 * KWIN];
    __shared__ float rVal[256];
    __shared__ int   rIdx[256];
    __shared__ int   rTid[256];

    const int b = blockIdx.x, tid = threadIdx.x;
    const float target = (float)KWIN / (float)NTOT;
    if (tid < COUT) sFac[tid] = __expf(target - duty[tid]);   // BOOST = 1.0
    __syncthreads();

    const float* base = pooled + (size_t)b * NTOT;

    float lv[KWIN];
    int   li[KWIN];
    #pragma unroll
    for (int j = 0; j < KWIN; ++j) { lv[j] = NEG_BIG; li[j] = 0; }

    for (int i = tid; i < NTOT; i += 256) {
        const float bv = base[i] * sFac[i / (HP * HP)];
        if (bv > lv[KWIN - 1]) {
            lv[KWIN - 1] = bv; li[KWIN - 1] = i;
            #pragma unroll
            for (int j = KWIN - 1; j > 0; --j) {
                if (lv[j] > lv[j - 1]) {
                    const float tv = lv[j]; lv[j] = lv[j - 1]; lv[j - 1] = tv;
                    const int   ti = li[j]; li[j] = li[j - 1]; li[j - 1] = ti;
                }
            }
        }
    }
    #pragma unroll
    for (int j = 0; j < KWIN; ++j) {
        sVal[tid * KWIN + j] = lv[j];
        sIdx[tid * KWIN + j] = li[j];
    }
    __syncthreads();

    int cur = 0;
    for (int rnd = 0; rnd < KWIN; ++rnd) {
        const bool ok = (cur < KWIN);
        rVal[tid] = ok ? sVal[tid * KWIN + cur] : NEG_BIG;
        rIdx[tid] = ok ? sIdx[tid * KWIN + cur] : 0;
        rTid[tid] = tid;
        __syncthreads();
        for (int s = 128; s > 0; s >>= 1) {
            if (tid < s && rVal[tid + s] > rVal[tid]) {
                rVal[tid] = rVal[tid + s];
                rIdx[tid] = rIdx[tid + s];
                rTid[tid] = rTid[tid + s];
            }
            __syncthreads();
        }
        if (tid == rTid[0]) cur++;                       // advance winner cursor
        if (tid == 0) {
            const int fi = rIdx[0];
            widx[b * KWIN + rnd] = fi;
            wval[b * KWIN + rnd] = base[fi];             // unboosted value
        }
        __syncthreads();
    }
}

// ---------------------------------------------------------------------------
// Kernel 3: sparse FC1 gather + ReLU. grid = 128, block = 512 (h index).
// Only 20 columns of the 504 MB weight matrix are touched per sample.
// ---------------------------------------------------------------------------
__global__ __launch_bounds__(512) void fc1_sparse(
    const float* __restrict__ fc1_w,    // [512, 246016]
    const float* __restrict__ fc1_b,    // [512]
    const int*   __restrict__ widx,
    const float* __restrict__ wval,
    float* __restrict__ h1)             // [B, 512]
{
    __shared__ int   sI[KWIN];
    __shared__ float sV[KWIN];
    const int b = blockIdx.x, h = threadIdx.x;
    if (h < KWIN) { sI[h] = widx[b * KWIN + h]; sV[h] = wval[b * KWIN + h]; }
    __syncthreads();

    const float* wrow = fc1_w + (size_t)h * NTOT;
    __builtin_prefetch(wrow + sI[0], 0, 1);              // global_prefetch_b8
    float acc = fc1_b[h];
    #pragma unroll
    for (int j = 0; j < KWIN; ++j) {
        if (j + 1 < KWIN) __builtin_prefetch(wrow + sI[j + 1], 0, 1);
        acc = fmaf(sV[j], wrow[sI[j]], acc);
    }
    h1[b * NHID + h] = fmaxf(acc, 0.0f);
}

// ---------------------------------------------------------------------------
// Kernel 4: FC2 + log_softmax. grid = 128, one wave32 per sample.
// ---------------------------------------------------------------------------
__global__ __launch_bounds__(32) void fc2_logsoftmax(
    const float* __restrict__ h1,
    const float* __restrict__ fc2_w,    // [10, 512]
    const float* __restrict__ fc2_b,    // [10]
    float* __restrict__ out)            // [B, 10]
{
    const int b = blockIdx.x, lane = threadIdx.x;

    float z[NCLS];
    #pragma unroll
    for (int c = 0; c < NCLS; ++c) z[c] = 0.0f;

    for (int h = lane; h < NHID; h += 32) {
        const float a = h1[b * NHID + h];
        #pragma unroll
        for (int c = 0; c < NCLS; ++c)
            z[c] = fmaf(a, fc2_w[c * NHID + h], z[c]);
    }
    #pragma unroll
    for (int c = 0; c < NCLS; ++c) {
        #pragma unroll
        for (int off = 16; off > 0; off >>= 1)
            z[c] += __shfl_xor(z[c], off, 32);
        z[c] += fc2_b[c];
    }

    float m = z[0];
    #pragma unroll
    for (int c = 1; c < NCLS; ++c) m = fmaxf(m, z[c]);
    float s = 0.0f;
    #pragma unroll
    for (int c = 0; c < NCLS; ++c) s += __expf(z[c] - m);
    const float lse = m + __logf(s);

    if (lane == 0) {
        #pragma unroll
        for (int c = 0; c < NCLS; ++c) out[b * NCLS + c] = z[c] - lse;
    }
}

// ---------------------------------------------------------------------------
extern "C" void kernel_launch(void* const* d_in, const int* in_sizes, int n_in,
                              void* d_out, int out_size, void* d_ws, size_t ws_size,
                              hipStream_t stream) {
    const float* x      = (const float*)d_in[0];
    const float* conv_w = (const float*)d_in[1];
    const float* conv_b = (const float*)d_in[2];
    const float* fc1_w  = (const float*)d_in[3];
    const float* fc1_b  = (const float*)d_in[4];
    const float* fc2_w  = (const float*)d_in[5];
    const float* fc2_b  = (const float*)d_in[6];
    const float* duty   = (const float*)d_in[7];
    float* out = (float*)d_out;

    // workspace carve-up (~126.2 MB total)
    float* pooled = (float*)d_ws;                               // B*64*62*62
    int*   widx   = (int*)(pooled + (size_t)BATCH * NTOT);      // B*20
    float* wval   = (float*)(widx + BATCH * KWIN);              // B*20
    float* h1     = wval + BATCH * KWIN;                        // B*512

    dim3 gConv(8, 31, BATCH);
    conv_pool_wmma<<<gConv, 256, 0, stream>>>(x, conv_w, conv_b, pooled);
    topk_boosted<<<BATCH, 256, 0, stream>>>(pooled, duty, widx, wval);
    fc1_sparse<<<BATCH, NHID, 0, stream>>>(fc1_w, fc1_b, widx, wval, h1);
    fc2_logsoftmax<<<BATCH, 32, 0, stream>>>(h1, fc2_w, fc2_b, out);
}